// DynamicMaskHead_37950331027990
// MI455X (gfx1250) — compile-verified
//
#include <hip/hip_runtime.h>

// DynamicMaskHead (CondInst): per-instance 8->8->8->1 MLP over 256x256 pixels.
// fp32 WMMA 16x16x4 (K=8 via two chained WMMAs); ds_swizzle SWAPX16 for the
// C-layout -> B-layout inter-layer shuffle; full-wave coalesced 32-pixel stores.

typedef float v2f __attribute__((ext_vector_type(2)));
typedef float v8f __attribute__((ext_vector_type(8)));

#define HW 65536      // 256*256 pixels
#define NPARAM 153    // 64 + 64 + 8 + 8 + 8 + 1

// lanes 0-15: take 'lo'; lanes 16-31: take hi_src's lane (l-16) value.
// ds_swizzle 0x401F = group-of-32, xor_mask=0x10 (SWAPX16), and_mask=0x1f.
static __device__ __forceinline__ float swz_hi_sel(int lane, float lo, float hi_src) {
  int s = __builtin_amdgcn_ds_swizzle(__float_as_int(hi_src), 0x401F);
  return (lane < 16) ? lo : __int_as_float(s);
}

__global__ __launch_bounds__(256)
void dynmask_wmma_kernel(const float* __restrict__ feats,   // [8][65536]
                         const float* __restrict__ params,  // [64][153]
                         float* __restrict__ out)           // [64][65536]
{
  const int lane  = threadIdx.x & 31;
  const int wave  = threadIdx.x >> 5;
  const int inst  = blockIdx.y;
  const int m     = lane & 15;        // M row (A) / pixel-in-tile (store)
  const int khalf = (lane >> 4) << 1; // K offset contributed by lane half: 0 or 2

  const float* P = params + inst * NPARAM;

  // ---- per-wave setup: W0/W1 into WMMA fp32 A layout (16x4: VGPR0 = K0|K2, VGPR1 = K1|K3)
  v2f A0a = {0.f, 0.f}, A0b = {0.f, 0.f};  // layer0, K=0..3 and K=4..7
  v2f A1a = {0.f, 0.f}, A1b = {0.f, 0.f};  // layer1
  if (m < 8) {                              // rows 8..15 stay zero (M padding)
    const float* w0 = P;
    const float* w1 = P + 64;
    A0a.x = w0[m * 8 + khalf    ];  A0a.y = w0[m * 8 + khalf + 1];
    A0b.x = w0[m * 8 + khalf + 4];  A0b.y = w0[m * 8 + khalf + 5];
    A1a.x = w1[m * 8 + khalf    ];  A1a.y = w1[m * 8 + khalf + 1];
    A1b.x = w1[m * 8 + khalf + 4];  A1b.y = w1[m * 8 + khalf + 5];
  }

  // Block-uniform scalars: biases, final weights -> SGPRs via scalar loads.
  float w2r[8], b0r[8], b1r[8];
#pragma unroll
  for (int r = 0; r < 8; ++r) {
    w2r[r] = P[128 + r];
    b0r[r] = P[136 + r];
    b1r[r] = P[144 + r];
  }
  const float b2 = P[152];

  // Per-lane feature row pointers for the four B-operand registers:
  // B reg pairs channels (0|2), (1|3), (4|6), (5|7) across lane halves.
  const float* fp0 = feats + (0 + khalf) * HW + m;
  const float* fp1 = feats + (1 + khalf) * HW + m;
  const float* fp2 = feats + (4 + khalf) * HW + m;
  const float* fp3 = feats + (5 + khalf) * HW + m;

  const int wgid = blockIdx.x * 8 + wave;   // 0..255 waves per instance
  float* outp = out + inst * HW;

#pragma unroll 1
  for (int i = 0; i < 8; ++i) {
    const int pix = wgid * 256 + i * 32;
    float res[2];
#pragma unroll
    for (int t = 0; t < 2; ++t) {
      const int p0 = pix + t * 16;

      // ---- layer 0: C(16 rows=out-ch, 16 cols=pixels) = W0 * X, K=8 via 2 WMMAs
      v2f B0 = { fp0[p0], fp1[p0] };      // K 0..3
      v2f B1 = { fp2[p0], fp3[p0] };      // K 4..7
      v8f c = {0.f, 0.f, 0.f, 0.f, 0.f, 0.f, 0.f, 0.f};
      c = __builtin_amdgcn_wmma_f32_16x16x4_f32(false, A0a, false, B0, (short)0, c, false, false);
      c = __builtin_amdgcn_wmma_f32_16x16x4_f32(false, A0b, false, B1, (short)0, c, false, false);
#pragma unroll
      for (int r = 0; r < 8; ++r) c[r] = fmaxf(c[r] + b0r[r], 0.f);

      // ---- C layout -> layer1 B layout: channel r lives in VGPR r lanes0-15;
      // K positions 2,3,6,7 need it in lanes16-31 -> SWAPX16 + select.
      v2f D0, D1;
      D0.x = swz_hi_sel(lane, c[0], c[2]);
      D0.y = swz_hi_sel(lane, c[1], c[3]);
      D1.x = swz_hi_sel(lane, c[4], c[6]);
      D1.y = swz_hi_sel(lane, c[5], c[7]);

      // ---- layer 1
      v8f e = {0.f, 0.f, 0.f, 0.f, 0.f, 0.f, 0.f, 0.f};
      e = __builtin_amdgcn_wmma_f32_16x16x4_f32(false, A1a, false, D0, (short)0, e, false, false);
      e = __builtin_amdgcn_wmma_f32_16x16x4_f32(false, A1b, false, D1, (short)0, e, false, false);

      // ---- layer 2 (1x8 dot) + sigmoid, valid in lanes 0-15
      float logit = b2;
#pragma unroll
      for (int r = 0; r < 8; ++r)
        logit = fmaf(w2r[r], fmaxf(e[r] + b1r[r], 0.f), logit);
      res[t] = 1.f / (1.f + __expf(-logit));
    }
    // merge two 16-pixel tiles into one full-wave contiguous 128B store
    outp[pix + lane] = swz_hi_sel(lane, res[0], res[1]);
  }
}

extern "C" void kernel_launch(void* const* d_in, const int* /*in_sizes*/, int /*n_in*/,
                              void* d_out, int /*out_size*/, void* /*d_ws*/, size_t /*ws_size*/,
                              hipStream_t stream) {
  const float* feats  = (const float*)d_in[0];  // mask_feats (1,8,256,256)
  const float* params = (const float*)d_in[1];  // mask_head_params (64,153)
  // d_in[2] det_bbox, d_in[3] fpn_levels: unused by the reference computation
  float* out = (float*)d_out;                   // (64,1,256,256)
  dim3 grid(32, 64);                            // 32 blocks/instance x 64 instances
  dynmask_wmma_kernel<<<grid, 256, 0, stream>>>(feats, params, out);
}